// LEFBlock_44616120271099
// MI455X (gfx1250) — compile-verified
//
#include <hip/hip_runtime.h>
#include <cstdint>

// ---------------------------------------------------------------------------
// LEF block: 4-point edge-replicated stencil + pointwise nonlinearity.
// Memory-bound (28 B/elem, ~2 FLOP/B, ~352 MB total -> ~15 us roofline
// at 23.3 TB/s). Data-path optimizations for MI455X:
//   * TDM (tensor_load_to_lds) DMAs the u/v halo tiles into LDS (TENSORcnt);
//     wave0 issues the u tile, wave1 the v tile (independent counters).
//   * u0/p/q are streamed with non-temporal loads BEFORE the block barrier so
//     their latency overlaps the tensor DMA.
//   * stencil reads come from LDS (ds_load), eliminating halo re-reads.
//   * outputs streamed with non-temporal stores (no L2 pollution).
// ---------------------------------------------------------------------------

#define EPS   1e-6f
#define NMAXF 0.85f

#define TH    16          // tile rows computed per block
#define TW    128         // tile cols computed per block
#define LDS_H 18          // TH + 2 halo
#define LDS_W 130         // TW + 2 halo
#define TILE_ELEMS (LDS_H * LDS_W)
#define KITER ((TH * TW) / 256)   // 8 elements per thread

#define IMG   512
#define PLANES (16 * 3)
#define TILES_PER_PLANE ((IMG / TH) * (IMG / TW))   // 32 * 4 = 128

typedef uint32_t u32;
typedef uint64_t u64;
typedef __attribute__((ext_vector_type(4))) u32 v4u;
typedef __attribute__((ext_vector_type(8))) int v8i;
typedef __attribute__((ext_vector_type(4))) int v4i;

#if __has_builtin(__builtin_amdgcn_tensor_load_to_lds)
#define USE_TDM 1
#else
#define USE_TDM 0
#endif

#if USE_TDM
// Issue one TDM load of an LDS_H x LDS_W f32 tile from a 512x512 f32 plane.
// gsrc points at the (clamped, always in-bounds) tile start inside the plane.
__device__ __forceinline__ void tdm_load_tile(const float* gsrc, u32 lds_byte_addr)
{
    const u64 ga = (u64)(uintptr_t)gsrc;

    // ---- D# group 0 (128 b): count=1 | lds_addr | global_addr | type=2 ----
    v4u g0;
    g0.x = 1u;                                             // count=1, user desc
    g0.y = lds_byte_addr;                                  // lds_addr (bytes)
    g0.z = (u32)ga;                                        // global_addr[31:0]
    g0.w = ((u32)(ga >> 32) & 0x01FFFFFFu) | (2u << 30);   // addr[56:32] | type=2

    // ---- D# group 1 (256 b) ------------------------------------------------
    v8i g1;
    g1[0] = (int)(2u << 16);                   // wg_mask=0, data_size=2 (4 B)
    g1[1] = (int)(((u32)IMG & 0xFFFFu) << 16); // tensor_dim0[15:0] = 512
    g1[2] = (int)((((u32)IMG >> 16) & 0xFFFFu)
                  | (((u32)IMG & 0xFFFFu) << 16));   // dim0 hi | dim1 lo
    g1[3] = (int)((((u32)IMG >> 16) & 0xFFFFu)
                  | ((u32)LDS_W << 16));             // dim1 hi | tile_dim0=130
    g1[4] = (int)LDS_H;                        // tile_dim1=18, tile_dim2=0
    g1[5] = (int)IMG;                          // tensor_dim0_stride = 512
    g1[6] = 0;                                 // stride hi | dim1_stride lo
    g1[7] = 0;

    const v4i gz4 = {0, 0, 0, 0};              // groups 2/3: unused (2D tensor)
    const v8i gz8 = {0, 0, 0, 0, 0, 0, 0, 0};  // 6-arg toolchain extension group
    __builtin_amdgcn_tensor_load_to_lds(g0, g1, gz4, gz4, gz8, /*cpol=*/0);
}
#endif

__global__ __launch_bounds__(256)
void lef_stencil_kernel(const float* __restrict__ u,  const float* __restrict__ v,
                        const float* __restrict__ u0, const float* __restrict__ p,
                        const float* __restrict__ q,
                        const float* __restrict__ s_alpha, const float* __restrict__ s_beta,
                        const float* __restrict__ s_kappa, const float* __restrict__ s_lam,
                        const float* __restrict__ s_omega,
                        float* __restrict__ uo, float* __restrict__ vo)
{
    __shared__ float smem[2 * TILE_ELEMS];
    float* su = smem;
    float* sv = smem + TILE_ELEMS;

    const u32 bid   = blockIdx.x;
    const u32 plane = bid >> 7;          // / 128 tiles-per-plane
    const u32 t     = bid & 127u;
    const u32 tr    = t >> 2;            // 0..31
    const u32 tc    = t & 3u;            // 0..3
    const int r0    = (int)(tr * TH);
    const int c0    = (int)(tc * TW);
    // Clamped DMA window start: always fully in-bounds; combined with the
    // per-element clamped neighbor indices below this reproduces edge-pad
    // semantics exactly.
    const int rs    = min(max(r0 - 1, 0), IMG - LDS_H);
    const int cs    = min(max(c0 - 1, 0), IMG - LDS_W);

    const size_t pbase = (size_t)plane * (size_t)(IMG * IMG);
    const float* up_ = u + pbase;
    const float* vp_ = v + pbase;

    const u32 wave = threadIdx.x >> 5;   // wave32 index within the block

#if USE_TDM
    if (wave == 0) {                     // wave 0 DMAs the u halo tile
        tdm_load_tile(up_ + (size_t)rs * IMG + cs, (u32)(uintptr_t)su);
    } else if (wave == 1) {              // wave 1 DMAs the v halo tile
        tdm_load_tile(vp_ + (size_t)rs * IMG + cs, (u32)(uintptr_t)sv);
    }
#else
    for (int i = (int)threadIdx.x; i < TILE_ELEMS; i += 256) {
        const int rr = i / LDS_W, cc = i % LDS_W;
        su[i] = up_[(size_t)(rs + rr) * IMG + (cs + cc)];
        sv[i] = vp_[(size_t)(rs + rr) * IMG + (cs + cc)];
    }
#endif

    // ---- Overlap: stream the center-only inputs while the DMA is in flight.
    float u0r[KITER], pr[KITER], qr[KITER];
#pragma unroll
    for (int k = 0; k < KITER; ++k) {
        const int idx = (int)threadIdx.x + k * 256;
        const size_t gidx = pbase + (size_t)(r0 + (idx >> 7)) * IMG + (c0 + (idx & 127));
        u0r[k] = __builtin_nontemporal_load(u0 + gidx);
        pr[k]  = __builtin_nontemporal_load(p  + gidx);
        qr[k]  = __builtin_nontemporal_load(q  + gidx);
    }

#if USE_TDM
    if (wave < 2) {
        __builtin_amdgcn_s_wait_tensorcnt(0);   // each issuing wave drains its DMA
    }
#endif
    __syncthreads();

    const float alpha = *s_alpha;
    const float beta  = *s_beta;
    const float kappa = *s_kappa;
    const float lam   = *s_lam;
    const float omega = *s_omega;
    const float inv_k2 = 1.0f / (kappa * kappa);

#pragma unroll
    for (int k = 0; k < KITER; ++k) {
        const int idx = (int)threadIdx.x + k * 256;
        const int r = idx >> 7;          // 0..15
        const int c = idx & 127;         // 0..127
        const int gr = r0 + r, gc = c0 + c;
        const int lr  = gr - rs,            lc  = gc - cs;
        const int lrU = max(gr - 1, 0) - rs;
        const int lrD = min(gr + 1, IMG - 1) - rs;
        const int lcL = max(gc - 1, 0) - cs;
        const int lcR = min(gc + 1, IMG - 1) - cs;

        const float uc = su[lr  * LDS_W + lc ];
        const float uu = su[lrU * LDS_W + lc ];
        const float ud = su[lrD * LDS_W + lc ];
        const float ul = su[lr  * LDS_W + lcL];
        const float ur = su[lr  * LDS_W + lcR];

        const float vc = sv[lr  * LDS_W + lc ];
        const float vu = sv[lrU * LDS_W + lc ];
        const float vd = sv[lrD * LDS_W + lc ];
        const float vl = sv[lr  * LDS_W + lcL];
        const float vr = sv[lr  * LDS_W + lcR];

        const float u0v = u0r[k];
        const float pv  = pr[k];
        const float qv  = qr[k];

        const float us = fmaxf(uc, EPS);
        const float vs = fmaxf(vc, EPS);
        // pow(x, a) with uniform exponent -> exp(a*log(x)), x > 0 guaranteed.
        const float Nu = fminf(pv * __expf(alpha * __logf(vs)), NMAXF);
        const float Nv = fminf(qv * __expf(beta  * __logf(us)), NMAXF);

        // ---- u channel ----
        float avg = 0.25f * (uu + ud + ul + ur);
        float gx  = 0.5f * (ur - ul);
        float gy  = 0.5f * (ud - uu);
        float ce  = __expf(-(gx * gx + gy * gy) * inv_k2);
        float sm  = avg + 0.25f * Nu;
        float cand = ce * sm + (1.0f - ce) * uc;
        const float un = (1.0f - omega) * uc + omega * ((1.0f - lam) * cand + lam * u0v);

        // ---- v channel ----
        avg = 0.25f * (vu + vd + vl + vr);
        gx  = 0.5f * (vr - vl);
        gy  = 0.5f * (vd - vu);
        ce  = __expf(-(gx * gx + gy * gy) * inv_k2);
        sm  = avg + 0.25f * Nv;
        cand = ce * sm + (1.0f - ce) * vc;
        const float vn = (1.0f - omega) * vc + omega * ((1.0f - lam) * cand + lam * u0v);

        const size_t gidx = pbase + (size_t)gr * IMG + gc;
        __builtin_nontemporal_store(un, uo + gidx);
        __builtin_nontemporal_store(vn, vo + gidx);
    }
}

extern "C" void kernel_launch(void* const* d_in, const int* in_sizes, int n_in,
                              void* d_out, int out_size, void* d_ws, size_t ws_size,
                              hipStream_t stream)
{
    (void)in_sizes; (void)n_in; (void)d_ws; (void)ws_size; (void)out_size;

    const float* u  = (const float*)d_in[0];
    const float* v  = (const float*)d_in[1];
    const float* u0 = (const float*)d_in[2];
    const float* p  = (const float*)d_in[3];
    const float* q  = (const float*)d_in[4];
    const float* al = (const float*)d_in[5];
    const float* be = (const float*)d_in[6];
    const float* ka = (const float*)d_in[7];
    const float* la = (const float*)d_in[8];
    const float* om = (const float*)d_in[9];

    float* uo = (float*)d_out;
    float* vo = uo + (size_t)PLANES * IMG * IMG;

    dim3 grid(PLANES * TILES_PER_PLANE);   // 48 * 128 = 6144 blocks
    dim3 block(256);                       // 8 wave32
    hipLaunchKernelGGL(lef_stencil_kernel, grid, block, 0, stream,
                       u, v, u0, p, q, al, be, ka, la, om, uo, vo);
}